// Model_68384469287346
// MI455X (gfx1250) — compile-verified
//
#include <hip/hip_runtime.h>
#include <math.h>

// ---------------- model constants ----------------
#define Bv      256
#define Lv      512
#define Nv      22
#define PREDv   64
#define DMv     512
#define DSTATEv 16
#define DCONVv  2
#define DTRANKv 32      // (512+15)/16
#define TFEATv  7
#define Tv      29      // N + TFEAT tokens
#define ROWSv   (Bv*Tv) // 7424

typedef __attribute__((ext_vector_type(2))) float v2f;
typedef __attribute__((ext_vector_type(8))) float v8f;

// =====================================================================
// WMMA f32 GEMM:  C[M,N] = act( A[M,K] @ W[N,K]^T + bias )  (+= optional)
// Wave -> 16x64 strip (4 accumulators, A fragment reused 4x).
// Block = 8 waves as 4(M) x 2(N-group) -> 64x128 block tile.
// Uses V_WMMA_F32_16X16X4_F32 (exact f32, K-step 4).
// =====================================================================
__global__ __launch_bounds__(256) void k_gemm_f32(
    const float* __restrict__ A, int lda,
    const float* __restrict__ W, int ldw,
    const float* __restrict__ bias,
    float* __restrict__ C, int ldc,
    int M, int N, int K, int act, int accum)
{
    const int lane  = threadIdx.x & 31;
    const int wv    = threadIdx.x >> 5;
    const int tileM = blockIdx.x * 64  + (wv & 3) * 16;
    const int tileN = blockIdx.y * 128 + (wv >> 2) * 64;
    if (tileM >= M || tileN >= N) return;           // wave-uniform exit

    const int lmod  = lane & 15;
    const int lhalf = lane >> 4;                    // 0: K0/K1, 1: K2/K3

    const float* __restrict__ arow  = A + (size_t)(tileM + lmod) * lda;
    const float* __restrict__ wrow0 = W + (size_t)(tileN +  0 + lmod) * ldw;
    const float* __restrict__ wrow1 = W + (size_t)(tileN + 16 + lmod) * ldw;
    const float* __restrict__ wrow2 = W + (size_t)(tileN + 32 + lmod) * ldw;
    const float* __restrict__ wrow3 = W + (size_t)(tileN + 48 + lmod) * ldw;

    v8f acc0 = {0.f,0.f,0.f,0.f,0.f,0.f,0.f,0.f};
    v8f acc1 = acc0, acc2 = acc0, acc3 = acc0;

    #pragma unroll 2
    for (int k = 0; k < K; k += 4) {
        if ((k & 63) == 0 && k + 64 < K)
            __builtin_prefetch(arow + k + 64, 0, 0);      // global_prefetch_b8
        const int ko = k + lhalf * 2;
        v2f a  = *(const v2f*)(arow  + ko);               // A 16x4 fragment
        v2f b0 = *(const v2f*)(wrow0 + ko);               // B 4x16 fragments (W^T)
        v2f b1 = *(const v2f*)(wrow1 + ko);
        v2f b2 = *(const v2f*)(wrow2 + ko);
        v2f b3 = *(const v2f*)(wrow3 + ko);
        acc0 = __builtin_amdgcn_wmma_f32_16x16x4_f32(false, a, false, b0, (short)0, acc0, false, false);
        acc1 = __builtin_amdgcn_wmma_f32_16x16x4_f32(false, a, false, b1, (short)0, acc1, false, false);
        acc2 = __builtin_amdgcn_wmma_f32_16x16x4_f32(false, a, false, b2, (short)0, acc2, false, false);
        acc3 = __builtin_amdgcn_wmma_f32_16x16x4_f32(false, a, false, b3, (short)0, acc3, false, false);
    }

    v8f accs[4] = {acc0, acc1, acc2, acc3};
    #pragma unroll
    for (int j = 0; j < 4; ++j) {
        const int ncol = tileN + j * 16 + lmod;
        const float bv = bias ? bias[ncol] : 0.f;
        #pragma unroll
        for (int r = 0; r < 8; ++r) {
            const int m = tileM + lhalf * 8 + r;          // C layout: vgpr r -> row
            const size_t idx = (size_t)m * ldc + ncol;
            float v = accs[j][r] + bv;
            if (accum) v += C[idx];
            if (act == 1)      v = fmaxf(v, 0.f);                              // relu
            else if (act == 2) v = fmaxf(v, 0.f) + log1pf(expf(-fabsf(v)));    // softplus
            C[idx] = v;
        }
    }
}

// =====================================================================
// fill_missing: one block per (b,n) series of length 512 in LDS
// =====================================================================
__global__ __launch_bounds__(256) void k_fill(const float* __restrict__ x,
                                              float* __restrict__ out)
{
    __shared__ float sv[Lv];
    __shared__ int   fl[Lv];
    __shared__ int   pidx[Lv];
    __shared__ int   nidx[Lv];
    __shared__ float med;
    __shared__ int   nval;

    const int bn = blockIdx.x, b = bn / Nv, n = bn % Nv;
    const int tid = threadIdx.x;

    for (int t = tid; t < Lv; t += 256) {
        float v = x[((size_t)b * Lv + t) * Nv + n];
        sv[t] = v;
        int valid = (v != -9999.0f) && !((fabsf(v) < 1e-8f) && (n >= 21));
        fl[t] = valid;
    }
    if (tid == 0) { med = 0.f; nval = 0; }
    __syncthreads();

    if (tid == 0) {
        int p = -1;
        for (int t = 0; t < Lv; ++t) { if (fl[t]) p = t; pidx[t] = p; }
        p = -1;
        for (int t = Lv - 1; t >= 0; --t) { if (fl[t]) p = t; nidx[t] = p; }
        int cnt = 0;
        for (int t = 0; t < Lv; ++t) cnt += fl[t];
        nval = cnt;
    }
    __syncthreads();

    const int k = max((nval - 1) / 2, 0);           // rank of median
    for (int t = tid; t < Lv; t += 256) {
        if (fl[t]) {
            int less = 0, leq = 0;
            const float vt = sv[t];
            for (int s = 0; s < Lv; ++s)
                if (fl[s]) { less += (sv[s] < vt); leq += (sv[s] <= vt); }
            if (less <= k && k < leq) med = vt;     // rank-select (ties benign)
        }
    }
    __syncthreads();

    for (int t = tid; t < Lv; t += 256) {
        float v;
        if (fl[t]) v = sv[t];
        else {
            const int pi = pidx[t], ni = nidx[t];
            if (pi >= 0 && ni >= 0) v = 0.5f * (sv[pi] + sv[ni]);
            else if (ni >= 0)       v = sv[ni];
            else if (pi >= 0)       v = sv[pi];
            else                    v = med;
        }
        out[((size_t)b * Lv + t) * Nv + n] = v;
    }
}

// =====================================================================
// Hampel (win 15, 3 sigma) + Savitzky-Golay (cubic, win 21) — channels 1..18
// SG center coefficients closed form: h_i = 3*(329 - 5 i^2)/9177
// =====================================================================
__global__ __launch_bounds__(256) void k_hampel_sg(const float* __restrict__ xf,
                                                   float* __restrict__ out)
{
    __shared__ float sv[Lv];
    __shared__ float xh[Lv];
    const int bn = blockIdx.x, b = bn / 18, n = 1 + bn % 18;
    const int tid = threadIdx.x;

    for (int t = tid; t < Lv; t += 256)
        sv[t] = xf[((size_t)b * Lv + t) * Nv + n];
    __syncthreads();

    for (int t = tid; t < Lv; t += 256) {
        float w[15], srt[15];
        #pragma unroll
        for (int i = 0; i < 15; ++i) {
            int idx = t + i - 7;
            idx = idx < 0 ? 0 : (idx > Lv - 1 ? Lv - 1 : idx);   // edge pad
            w[i] = sv[idx];
        }
        #pragma unroll
        for (int i = 0; i < 15; ++i) {                            // insertion sort
            float v = w[i]; int j = i;
            while (j > 0 && srt[j - 1] > v) { srt[j] = srt[j - 1]; --j; }
            srt[j] = v;
        }
        const float md = srt[7];
        #pragma unroll
        for (int i = 0; i < 15; ++i) {
            float v = fabsf(w[i] - md); int j = i;
            while (j > 0 && srt[j - 1] > v) { srt[j] = srt[j - 1]; --j; }
            srt[j] = v;
        }
        const float mad = srt[7];
        const float xv = sv[t];
        xh[t] = (fabsf(xv - md) > 1.4826f * mad * 3.0f) ? md : xv;
    }
    __syncthreads();

    for (int t = tid; t < Lv; t += 256) {
        float acc = 0.f;
        #pragma unroll
        for (int i = -10; i <= 10; ++i) {
            int idx = t + i;
            if (idx < 0)      idx = -idx;                          // reflect pad
            if (idx > Lv - 1) idx = 2 * (Lv - 1) - idx;
            const float hc = 3.0f * (329.0f - 5.0f * (float)(i * i)) / 9177.0f;
            acc += hc * xh[idx];
        }
        out[((size_t)b * Lv + t) * Nv + n] = acc;
    }
}

// channels 19,20,21 pass through the fill_missing output unchanged
__global__ void k_copy3(const float* __restrict__ xf, float* __restrict__ out)
{
    const int gid = blockIdx.x * blockDim.x + threadIdx.x;
    if (gid >= Bv * Lv * 3) return;
    const int c = 19 + gid % 3;
    const int t = (gid / 3) % Lv;
    const int b = gid / (3 * Lv);
    const size_t idx = ((size_t)b * Lv + t) * Nv + c;
    out[idx] = xf[idx];
}

// gaussian_diffuse(sigma=7, radius=21) on channel 0, reflect pad
__global__ __launch_bounds__(256) void k_gauss(const float* __restrict__ xf,
                                               float* __restrict__ out)
{
    __shared__ float sv[Lv];
    __shared__ float kw[43];
    __shared__ float ksum;
    const int b = blockIdx.x, tid = threadIdx.x;

    for (int t = tid; t < Lv; t += 256)
        sv[t] = xf[((size_t)b * Lv + t) * Nv + 0];
    if (tid < 43) {
        const float d = ((float)tid - 21.f) / 7.f;
        kw[tid] = expf(-0.5f * d * d);
    }
    __syncthreads();
    if (tid == 0) {
        float s = 0.f;
        for (int i = 0; i < 43; ++i) s += kw[i];
        ksum = fmaxf(s, 1e-8f);
    }
    __syncthreads();

    for (int t = tid; t < Lv; t += 256) {
        float acc = 0.f;
        for (int i = -21; i <= 21; ++i) {
            int idx = t + i;
            if (idx < 0)      idx = -idx;
            if (idx > Lv - 1) idx = 2 * (Lv - 1) - idx;
            acc += kw[i + 21] * sv[idx];
        }
        out[((size_t)b * Lv + t) * Nv + 0] = acc / ksum;
    }
}

// tok[(b*29+k)][t] : k<22 -> xs[b,t,k] ; else x_mark[b,t,k-22]
__global__ void k_tok(const float* __restrict__ xs,
                      const float* __restrict__ xmark,
                      float* __restrict__ tok)
{
    const int r = blockIdx.x, b = r / Tv, k = r % Tv;
    for (int t = threadIdx.x; t < Lv; t += 256) {
        float v;
        if (k < Nv) v = xs[((size_t)b * Lv + t) * Nv + k];
        else        v = xmark[((size_t)b * Lv + t) * TFEATv + (k - Nv)];
        tok[(size_t)r * Lv + t] = v;
    }
}

// depthwise causal conv (K=2) + silu; rev flips the neighbor direction
__global__ void k_conv_silu(const float* __restrict__ XZ,
                            const float* __restrict__ convW,
                            const float* __restrict__ convb,
                            float* __restrict__ XC, int rev)
{
    const int gid = blockIdx.x * blockDim.x + threadIdx.x;
    if (gid >= ROWSv * DMv) return;
    const int c  = gid & (DMv - 1);
    const int r  = gid >> 9;
    const int tk = r % Tv;
    const float cur = XZ[(size_t)r * 1024 + c];
    float prev = 0.f;
    if (!rev) { if (tk > 0)      prev = XZ[(size_t)(r - 1) * 1024 + c]; }
    else      { if (tk < Tv - 1) prev = XZ[(size_t)(r + 1) * 1024 + c]; }
    const float v = convW[c * 2 + 0] * prev + convW[c * 2 + 1] * cur + convb[c];
    XC[gid] = v / (1.f + expf(-v));
}

// selective-scan: each thread owns one (b,c), 16 states in registers, 29 steps
__global__ __launch_bounds__(256) void k_scan(
    const float* __restrict__ DT,   // ROWS x 512
    const float* __restrict__ XC,   // ROWS x 512
    const float* __restrict__ XDB,  // ROWS x 64 (B at +32, C at +48)
    const float* __restrict__ XZ,   // ROWS x 1024 (z at +512)
    const float* __restrict__ Alog, // 512 x 16
    const float* __restrict__ Dp,   // 512
    float* __restrict__ Y, int rev)
{
    const int gid = blockIdx.x * 256 + threadIdx.x;
    const int b = gid >> 9, c = gid & (DMv - 1);

    float Ar[DSTATEv], h[DSTATEv];
    #pragma unroll
    for (int s = 0; s < DSTATEv; ++s) { Ar[s] = -expf(Alog[c * DSTATEv + s]); h[s] = 0.f; }
    const float Dv = Dp[c];

    for (int t = 0; t < Tv; ++t) {
        const int tk = rev ? (Tv - 1 - t) : t;
        const size_t r = (size_t)b * Tv + tk;
        const float dtv = DT[r * DMv + c];
        const float xcv = XC[r * DMv + c];
        const float dx  = dtv * xcv;
        const float* __restrict__ bm = XDB + r * 64 + 32;
        const float* __restrict__ cm = XDB + r * 64 + 48;
        float yv = 0.f;
        #pragma unroll
        for (int s = 0; s < DSTATEv; ++s) {
            h[s] = expf(dtv * Ar[s]) * h[s] + dx * bm[s];
            yv += h[s] * cm[s];
        }
        yv += Dv * xcv;
        const float zv = XZ[r * 1024 + DMv + c];
        yv *= zv / (1.f + expf(-zv));               // silu gate
        Y[r * DMv + c] = yv;
    }
}

// O = layernorm(X (+R)) * w + b, row length 512, one block per row
__global__ __launch_bounds__(256) void k_add_ln(
    const float* __restrict__ X, const float* __restrict__ R,
    const float* __restrict__ w, const float* __restrict__ bb,
    float* __restrict__ O)
{
    __shared__ float red[256];
    __shared__ float s_mean;
    const int r = blockIdx.x, tid = threadIdx.x;
    float v0 = X[(size_t)r * DMv + tid];
    float v1 = X[(size_t)r * DMv + tid + 256];
    if (R) { v0 += R[(size_t)r * DMv + tid]; v1 += R[(size_t)r * DMv + tid + 256]; }

    red[tid] = v0 + v1;
    __syncthreads();
    for (int s = 128; s > 0; s >>= 1) { if (tid < s) red[tid] += red[tid + s]; __syncthreads(); }
    if (tid == 0) s_mean = red[0] / (float)DMv;
    __syncthreads();
    const float mean = s_mean;
    const float d0 = v0 - mean, d1 = v1 - mean;
    __syncthreads();
    red[tid] = d0 * d0 + d1 * d1;
    __syncthreads();
    for (int s = 128; s > 0; s >>= 1) { if (tid < s) red[tid] += red[tid + s]; __syncthreads(); }
    const float inv = rsqrtf(red[0] / (float)DMv + 1e-5f);
    O[(size_t)r * DMv + tid]       = d0 * inv * w[tid]       + bb[tid];
    O[(size_t)r * DMv + tid + 256] = d1 * inv * w[tid + 256] + bb[tid + 256];
}

// out[b,i,j] = proj[(b*29 + j)*64 + i], i<64, j<22
__global__ void k_out(const float* __restrict__ prj, float* __restrict__ out)
{
    const int gid = blockIdx.x * blockDim.x + threadIdx.x;
    if (gid >= Bv * PREDv * Nv) return;
    const int j = gid % Nv;
    const int i = (gid / Nv) % PREDv;
    const int b = gid / (Nv * PREDv);
    out[gid] = prj[((size_t)b * Tv + j) * PREDv + i];
}

// =====================================================================
static inline void gemm(hipStream_t st, const float* A, int lda,
                        const float* W, int ldw, const float* bias,
                        float* C, int ldc, int M, int N, int K,
                        int act, int accum)
{
    dim3 g((M + 63) / 64, (N + 127) / 128);
    k_gemm_f32<<<g, 256, 0, st>>>(A, lda, W, ldw, bias, C, ldc, M, N, K, act, accum);
}

extern "C" void kernel_launch(void* const* d_in, const int* in_sizes, int n_in,
                              void* d_out, int out_size, void* d_ws, size_t ws_size,
                              hipStream_t stream)
{
    const float* x_enc  = (const float*)d_in[0];
    const float* x_mark = (const float*)d_in[1];
    auto P = [&](int i) { return (const float*)d_in[i]; };
    // params: 4 emb_W, 5 emb_b; per layer l base=6+l*26:
    //   +0..8  f: Win convW convb Wx Wdt bdt Alog D Wout
    //   +9..17 r: same
    //   +18 ffn1W +19 ffn1b +20 ffn2W +21 ffn2b +22 ln1w +23 ln1b +24 ln2w +25 ln2b
    // 58 normw, 59 normb, 60 projW, 61 projb

    float* ws = (float*)d_ws;
    size_t o = 0;
    float* XSF = ws + o; o += (size_t)Bv * Lv * Nv;     // fill_missing out
    float* XSP = ws + o; o += (size_t)Bv * Lv * Nv;     // preprocessed
    float* H   = ws + o; o += (size_t)ROWSv * DMv;      // hidden state
    float* XZ  = ws + o; o += (size_t)ROWSv * 1024;     // Win output (xi|z)
    float* XC  = ws + o; o += (size_t)ROWSv * DMv;      // conv+silu
    float* XDB = ws + o; o += (size_t)ROWSv * 64;       // dt_in|B|C ; reused for proj
    float* DT  = ws + o; o += (size_t)ROWSv * DMv;      // softplus dt
    float* Y   = ws + o; o += (size_t)ROWSv * DMv;      // scan out ; reused for tok
    float* NX  = ws + o; o += (size_t)ROWSv * DMv;      // mamba_f + mamba_r / ffn2
    float* FF  = ws + o; o += (size_t)ROWSv * DMv;      // ffn1 out
    (void)in_sizes; (void)n_in; (void)out_size; (void)ws_size;

    // ---- preprocessing ----
    k_fill<<<Bv * Nv, 256, 0, stream>>>(x_enc, XSF);
    k_hampel_sg<<<Bv * 18, 256, 0, stream>>>(XSF, XSP);
    k_copy3<<<(Bv * Lv * 3 + 255) / 256, 256, 0, stream>>>(XSF, XSP);
    k_gauss<<<Bv, 256, 0, stream>>>(XSF, XSP);

    // ---- embedding ----
    k_tok<<<ROWSv, 256, 0, stream>>>(XSP, x_mark, Y);
    gemm(stream, Y, Lv, P(4), Lv, P(5), H, DMv, ROWSv, DMv, Lv, 0, 0);

    // ---- encoder layers ----
    for (int l = 0; l < 2; ++l) {
        const int lb = 6 + l * 26;
        for (int dir = 0; dir < 2; ++dir) {
            const int pb = lb + dir * 9;
            const float* Win   = P(pb + 0);
            const float* convW = P(pb + 1);
            const float* convb = P(pb + 2);
            const float* Wx    = P(pb + 3);
            const float* Wdt   = P(pb + 4);
            const float* bdt   = P(pb + 5);
            const float* Alog  = P(pb + 6);
            const float* Dp    = P(pb + 7);
            const float* Wout  = P(pb + 8);

            gemm(stream, H, DMv, Win, DMv, nullptr, XZ, 1024, ROWSv, 1024, DMv, 0, 0);
            k_conv_silu<<<(ROWSv * DMv) / 256, 256, 0, stream>>>(XZ, convW, convb, XC, dir);
            gemm(stream, XC, DMv, Wx, DMv, nullptr, XDB, 64, ROWSv, 64, DMv, 0, 0);
            gemm(stream, XDB, 64, Wdt, DTRANKv, bdt, DT, DMv, ROWSv, DMv, DTRANKv, 2, 0);
            k_scan<<<(Bv * DMv) / 256, 256, 0, stream>>>(DT, XC, XDB, XZ, Alog, Dp, Y, dir);
            gemm(stream, Y, DMv, Wout, DMv, nullptr, NX, DMv, ROWSv, DMv, DMv, 0, dir);
        }
        k_add_ln<<<ROWSv, 256, 0, stream>>>(H, NX, P(lb + 22), P(lb + 23), H);
        gemm(stream, H, DMv, P(lb + 18), DMv, P(lb + 19), FF, DMv, ROWSv, DMv, DMv, 1, 0);
        gemm(stream, FF, DMv, P(lb + 20), DMv, P(lb + 21), NX, DMv, ROWSv, DMv, DMv, 0, 0);
        k_add_ln<<<ROWSv, 256, 0, stream>>>(H, NX, P(lb + 24), P(lb + 25), H);
    }

    // ---- final norm + projection + gather ----
    k_add_ln<<<ROWSv, 256, 0, stream>>>(H, nullptr, P(58), P(59), H);
    gemm(stream, H, DMv, P(60), DMv, P(61), XDB, PREDv, ROWSv, PREDv, DMv, 0, 0);
    k_out<<<(Bv * PREDv * Nv + 255) / 256, 256, 0, stream>>>(XDB, (float*)d_out);
}